// LaplaceNet_69801808494799
// MI455X (gfx1250) — compile-verified
//
#include <hip/hip_runtime.h>
#include <stdint.h>

// Screened-Laplace Jacobi (10 iters) on 2048^2 for MI455X (gfx1250).
// Working set (~64MB: 2 padded V buffers + fused A/B coefficients) stays in
// the 192MB L2; each step is an L2-streaming stencil pass. Halo tiles are
// DMA'd into LDS with the CDNA5 Tensor Data Mover (tensor_load_to_lds /
// TENSORcnt), and all per-lane traffic is 128-bit.

#define RES  2048
#define PADW 2056           // padded pitch; interior col j at index j+XOFF
#define PADH 2050           // padded height; interior row i at index i+1
#define XOFF 4              // left alignment pad (ring col at XOFF-1 = 3)
#define TX   64
#define TY   16
#define LTX  72             // LDS tile row pitch (dwords, mult of 4 -> 16B rows)
#define LTY  18

typedef __attribute__((ext_vector_type(4))) uint32_t u32x4;
typedef __attribute__((ext_vector_type(8))) uint32_t u32x8;
typedef __attribute__((ext_vector_type(4))) float    f32x4;
typedef __attribute__((ext_vector_type(4))) int      i32x4;

// ---------------------------------------------------------------------------
// Pass 0 (HBM-bound, fully vectorized): fuse mask/bc/C into affine coeffs,
//   A = mask ? 0 : 1/(4+relu(C)),  B = mask ? bc : 0   =>  V_new = nsum*A + B
// emit C_pos (output 1), and build the padded V0 with replicated edge ring.
// ---------------------------------------------------------------------------
__global__ __launch_bounds__(256)
void lap_init(const float* __restrict__ x, const float* __restrict__ C,
              const int* __restrict__ bt, const float* __restrict__ bc,
              float* __restrict__ Vp, float* __restrict__ A,
              float* __restrict__ B, float* __restrict__ Cpos)
{
    int g = blockIdx.x * blockDim.x + threadIdx.x;   // one float4 group each
    if (g >= RES * (RES / 4)) return;
    const int i   = g >> 9;                          // / (RES/4)
    const int j0  = (g & 511) << 2;
    const int idx = i * RES + j0;

    // unconditional wide loads: no exec juggling
    f32x4 xv  = *(const f32x4*)&x[idx];
    f32x4 cv  = *(const f32x4*)&C[idx];
    f32x4 bcv = *(const f32x4*)&bc[idx];
    i32x4 btv = *(const i32x4*)&bt[idx];

    f32x4 v0, a, b, cp;
#pragma unroll
    for (int k = 0; k < 4; ++k) {
        bool m = btv[k] > 0;
        cp[k]  = fmaxf(cv[k], 0.0f);
        v0[k]  = m ? bcv[k] : xv[k];
        a[k]   = m ? 0.0f : 1.0f / (4.0f + cp[k]);
        b[k]   = m ? bcv[k] : 0.0f;
    }

    __builtin_nontemporal_store(cp, (f32x4*)&Cpos[idx]);   // write-once output
    *(f32x4*)&A[idx] = a;                                  // L2-resident
    *(f32x4*)&B[idx] = b;

    const size_t pbase = (size_t)(i + 1) * PADW + (j0 + XOFF);
    *(f32x4*)&Vp[pbase] = v0;
    // replicate-edge ring (zero-Neumann outer boundary)
    if (i == 0)         *(f32x4*)&Vp[(size_t)0 * PADW + (j0 + XOFF)]         = v0;
    if (i == RES - 1)   *(f32x4*)&Vp[(size_t)(RES + 1) * PADW + (j0 + XOFF)] = v0;
    if (j0 == 0)            Vp[pbase - 1] = v0[0];
    if (j0 + 3 == RES - 1)  Vp[pbase + 4] = v0[3];
}

// ---------------------------------------------------------------------------
// One Jacobi step. blockDim = (16,16) = 8 wave32; block owns a 64x16 tile.
// Wave 0 issues one TDM descriptor DMAing the 72x18 (aligned, halo-inclusive)
// tile into LDS, waits TENSORcnt, barrier, then all threads stencil 4
// consecutive cells with aligned b128 LDS/global traffic.
// ---------------------------------------------------------------------------
__global__ __launch_bounds__(256)
void lap_step(const float* __restrict__ Vsrc, float* __restrict__ Vdst,
              const float* __restrict__ A, const float* __restrict__ B,
              float* __restrict__ out, int final_step)
{
    __shared__ float tile[LTY * LTX];                // 5184 B

    const int bx = blockIdx.x, by = blockIdx.y;
    const int tx = threadIdx.x, ty = threadIdx.y;    // (16,16)
    const uint32_t tid = (uint32_t)(tx + ty * 16);

    // Scalar (SCC) branch: TDM ignores EXEC, so gate at wave granularity.
    if (__builtin_amdgcn_readfirstlane((int)tid) == 0) {
        // Tile starts at padded (by*TY, bx*TX); interiors land 16B-aligned.
        uint64_t ga = (uint64_t)(uintptr_t)Vsrc +
                      4ull * ((uint64_t)(by * TY) * PADW + (uint64_t)(bx * TX));
        uint32_t lds = (uint32_t)(uintptr_t)&tile[0];  // lo32 flat = LDS byte addr

        u32x4 g0;                                    // D# group 0 (ISA 8.3)
        g0[0] = 1u;                                  // count=1, user mode
        g0[1] = lds;                                 // lds_addr
        g0[2] = (uint32_t)ga;                        // global_addr[31:0]
        g0[3] = (uint32_t)((ga >> 32) & 0x01FFFFFFu) | (2u << 30); // ga hi | type=2

        u32x8 g1;                                    // D# group 1 (ISA 8.4)
        g1[0] = 2u << 16;                            // data_size=4B
        g1[1] = (uint32_t)PADW << 16;                // tensor_dim0 lo16 (2056)
        g1[2] = (uint32_t)PADH << 16;                // dim0 hi=0 | tensor_dim1 lo16
        g1[3] = (uint32_t)LTX << 16;                 // dim1 hi=0 | tile_dim0 = 72
        g1[4] = (uint32_t)LTY;                       // tile_dim1 = 18 | tile_dim2 = 0
        g1[5] = (uint32_t)PADW;                      // tensor_dim0_stride lo32
        g1[6] = 0u;                                  // stride0 hi | stride1 lo
        g1[7] = 0u;                                  // stride1 hi

        asm volatile("tensor_load_to_lds %0, %1" :: "s"(g0), "s"(g1) : "memory");
        __builtin_amdgcn_s_wait_tensorcnt(0);
    }
    __syncthreads();

    const int c0    = tx << 2;                       // local col of first cell
    const int j0    = bx * TX + c0;
    const int i     = by * TY + ty;
    const int lbase = (ty + 1) * LTX + XOFF + c0;    // 16B-aligned

    f32x4 up  = *(const f32x4*)&tile[lbase - LTX];
    f32x4 dn  = *(const f32x4*)&tile[lbase + LTX];
    f32x4 cen = *(const f32x4*)&tile[lbase];
    float lm  = tile[lbase - 1];
    float rp  = tile[lbase + 4];

    f32x4 nsum;
    nsum[0] = (up[0] + dn[0]) + (lm     + cen[1]);
    nsum[1] = (up[1] + dn[1]) + (cen[0] + cen[2]);
    nsum[2] = (up[2] + dn[2]) + (cen[1] + cen[3]);
    nsum[3] = (up[3] + dn[3]) + (cen[2] + rp);

    const int idx = i * RES + j0;
    f32x4 a = *(const f32x4*)&A[idx];
    f32x4 b = *(const f32x4*)&B[idx];
    f32x4 v;
#pragma unroll
    for (int k = 0; k < 4; ++k) v[k] = fmaf(nsum[k], a[k], b[k]);

    if (final_step) {
        f32x4 o;
#pragma unroll
        for (int k = 0; k < 4; ++k) o[k] = (v[k] >= 1.0f) ? 0.95f : v[k];
        __builtin_nontemporal_store(o, (f32x4*)&out[idx]);  // bc<1: mask cells ok
    } else {
        const size_t pbase = (size_t)(i + 1) * PADW + (j0 + XOFF);
        *(f32x4*)&Vdst[pbase] = v;
        if (i == 0)         *(f32x4*)&Vdst[(size_t)0 * PADW + (j0 + XOFF)]         = v;
        if (i == RES - 1)   *(f32x4*)&Vdst[(size_t)(RES + 1) * PADW + (j0 + XOFF)] = v;
        if (j0 == 0)            Vdst[pbase - 1] = v[0];
        if (j0 + 3 == RES - 1)  Vdst[pbase + 4] = v[3];
    }
}

// ---------------------------------------------------------------------------
extern "C" void kernel_launch(void* const* d_in, const int* in_sizes, int n_in,
                              void* d_out, int out_size, void* d_ws, size_t ws_size,
                              hipStream_t stream)
{
    (void)in_sizes; (void)n_in; (void)out_size; (void)ws_size;

    const float* x  = (const float*)d_in[0];
    const float* C  = (const float*)d_in[1];
    const int*   bt = (const int*)d_in[2];
    const float* bc = (const float*)d_in[3];

    float* out  = (float*)d_out;                 // output 0: V (RES*RES)
    float* Cpos = out + (size_t)RES * RES;       // output 1: relu(C)

    const size_t PP = (size_t)PADW * PADH;
    float* ws    = (float*)d_ws;
    float* Vping = ws;                           // padded potential, buffer A
    float* Vpong = ws + PP;                      // padded potential, buffer B
    float* Acoef = ws + 2 * PP;                  // 1/(4+Cpos) or 0
    float* Bcoef = Acoef + (size_t)RES * RES;    // bc or 0

    lap_init<<<(RES * (RES / 4)) / 256, 256, 0, stream>>>(x, C, bt, bc,
                                                          Vping, Acoef, Bcoef, Cpos);

    dim3 grid(RES / TX, RES / TY);               // 32 x 128 blocks
    dim3 block(16, 16);                          // 256 threads = 8 wave32
    for (int k = 0; k < 10; ++k) {
        const float* src = (k & 1) ? Vpong : Vping;
        float*       dst = (k & 1) ? Vping : Vpong;
        lap_step<<<grid, block, 0, stream>>>(src, dst, Acoef, Bcoef, out, k == 9);
    }
}